// Simplified_MultiHeadAttention_50878182588810
// MI455X (gfx1250) — compile-verified
//
#include <hip/hip_runtime.h>
#include <math.h>

#define B_    4
#define T_    2048
#define D_    1024
#define H_    8
#define NT_   128      // full 16-row tiles covering T
#define NTIL_ 129      // output tiles over T+1 = 2049 rows
#define PITCH 1032     // LDS row pitch in floats (bank-spread, 16B-aligned rows)
#define EPS_  1e-7f

typedef __attribute__((ext_vector_type(2))) float v2f;
typedef __attribute__((ext_vector_type(4))) float f4;
typedef __attribute__((ext_vector_type(8))) float v8f;

// ---------------- Kernel 1: per-16-row-tile column sums ----------------
__global__ void __launch_bounds__(256) k_blocksum(const float* __restrict__ X,
                                                  float* __restrict__ bs) {
    const int b  = blockIdx.x >> 7;       // 4 batches
    const int it = blockIdx.x & 127;      // 128 tiles
    const int t  = threadIdx.x;           // 256 threads -> float4 columns
    const f4* X4 = (const f4*)X;
    f4 acc = (f4){0.f, 0.f, 0.f, 0.f};
    const int base = (b * T_ + it * 16) * (D_ / 4) + t;
#pragma unroll
    for (int r = 0; r < 16; ++r) acc += X4[base + r * (D_ / 4)];
    ((f4*)bs)[(b * NT_ + it) * (D_ / 4) + t] = acc;
}

// ---------------- Kernel 2: exclusive scan of tile sums -> carries ----------------
__global__ void __launch_bounds__(256) k_carry(const float* __restrict__ bs,
                                               float* __restrict__ carry) {
    const int b = blockIdx.x >> 2;
    const int c = (blockIdx.x & 3) * 256 + threadIdx.x;
    float run = 0.f;
    for (int it = 0; it < NT_; ++it) {
        carry[(size_t)(b * NTIL_ + it) * D_ + c] = run;
        run += bs[(size_t)(b * NT_ + it) * D_ + c];
    }
    carry[(size_t)(b * NTIL_ + NT_) * D_ + c] = run;   // total sum (used by row i == T)
}

// ---------------- Kernel 3: WMMA scan + softmax-structured V + norm + write ----------------
__global__ void __launch_bounds__(256) k_main(const float* __restrict__ X,
                                              const float* __restrict__ W,
                                              const float* __restrict__ carry,
                                              float* __restrict__ out) {
    const int itile = blockIdx.x;          // 0..128
    const int b     = blockIdx.y;          // 0..3
    const int t0    = itile * 16;
    const int tid   = threadIdx.x;

    extern __shared__ float smem[];
    float* xw  = smem;                 // 16 x PITCH : X rows t0..t0+15 (0 if >= T)
    float* Sl  = xw  + 16 * PITCH;     // 16 x PITCH : inclusive prefix sums (+carry)
    float* x0r = Sl  + 16 * PITCH;     // D_         : X[b,0,:] (row i==T correction)
    float* red = x0r + D_;             // 256        : reduction scratch
    float* inv = red + 256;            // 128        : 1/(||V||+eps) per (i_local,h)

    // softmax structure constant: e^(w_plus - w_minus)
    const float ew = __expf(W[0] - W[H_]);   // W[0,0]=w_plus, W[1,0]=w_minus

    // ---- Phase A: stage tile rows + row 0 into LDS (float4, coalesced) ----
    {
        const f4* X4 = (const f4*)X;
        const f4 z = (f4){0.f, 0.f, 0.f, 0.f};
#pragma unroll
        for (int j = 0; j < 16; ++j) {
            const int gr = t0 + j;
            f4 v = (gr < T_) ? X4[(size_t)(b * T_ + gr) * (D_ / 4) + tid] : z;
            *(f4*)&xw[j * PITCH + 4 * tid] = v;
        }
        *(f4*)&x0r[4 * tid] = X4[(size_t)(b * T_) * (D_ / 4) + tid];
    }
    __syncthreads();

    // ---- Phase B: triangular-ones matmul (inclusive scan) on the matrix pipe ----
    {
        const int wave = tid >> 5, lane = tid & 31;
        const int half = lane >> 4, l = lane & 15;
        // A = L16 (lower-triangular ones), split into four 16x4 K-chunks.
        v2f a[4];
#pragma unroll
        for (int kc = 0; kc < 4; ++kc) {
            const int k0 = kc * 4 + 2 * half;
            a[kc].x = (l >= k0)     ? 1.f : 0.f;
            a[kc].y = (l >= k0 + 1) ? 1.f : 0.f;
        }
#pragma unroll
        for (int q = 0; q < 8; ++q) {               // 8 d-blocks per wave, 64 total
            const int db  = wave * 8 + q;
            const int col = db * 16 + l;
            v8f c = (v8f){0.f, 0.f, 0.f, 0.f, 0.f, 0.f, 0.f, 0.f};
#pragma unroll
            for (int kc = 0; kc < 4; ++kc) {
                const int r0 = kc * 4 + 2 * half;   // B rows for this lane-half
                v2f bb;
                bb.x = xw[r0 * PITCH + col];
                bb.y = xw[(r0 + 1) * PITCH + col];
                c = __builtin_amdgcn_wmma_f32_16x16x4_f32(
                        false, a[kc], false, bb, (short)0, c, false, false);
            }
            const float cw = carry[(size_t)(b * NTIL_ + itile) * D_ + col];
#pragma unroll
            for (int r = 0; r < 8; ++r)
                Sl[(r + 8 * half) * PITCH + col] = c[r] + cw;
        }
    }
    __syncthreads();

    // ---- Phase C: ||V||^2 per (i_local, h): 128 tasks x 2 threads ----
    {
        const int task = tid >> 1, hc = tid & 1;
        const int il = task >> 3, h = task & 7;
        const int gi = t0 + il;
        float acc = 0.f;
        if (gi <= T_) {
            const bool last = (gi == T_);
            const int  n    = (gi + 1 < T_) ? (gi + 1) : T_;
            const float den = ew + (float)(n - 1);
            const float al  = 1.f / den, be = (ew - 1.f) / den;
            const float iu  = 1.f / (float)n;
            const bool plus = (h <= gi);
            int gr = gi - h; if (gr < 0) gr = 0;
            const f4* xrow = (gr >= t0) ? (const f4*)&xw[(gr - t0) * PITCH]
                                        : (const f4*)&X[(size_t)(b * T_ + gr) * D_];
            for (int c = hc * 512; c < hc * 512 + 512; c += 4) {
                f4 s = *(const f4*)&Sl[il * PITCH + c];
                if (last) s -= *(const f4*)&x0r[c];
                f4 v = plus ? (al * s + be * xrow[c >> 2]) : (s * iu);
                acc += v.x * v.x + v.y * v.y + v.z * v.z + v.w * v.w;
            }
        }
        red[tid] = acc;
    }
    __syncthreads();
    if (tid < 128) {
        const float tot = red[2 * tid] + red[2 * tid + 1];
        inv[tid] = 1.f / (sqrtf(tot) + EPS_);
    }
    __syncthreads();

    // ---- Phase D: write (B, T+1, D, 9); lanes -> consecutive columns ----
    for (int il = 0; il < 16; ++il) {
        const int gi = t0 + il;
        if (gi > T_) break;                       // uniform across the workgroup
        const bool last = (gi == T_);
        const int  n    = (gi + 1 < T_) ? (gi + 1) : T_;
        const float den = ew + (float)(n - 1);
        const float al  = 1.f / den, be = (ew - 1.f) / den;
        const float iu  = 1.f / (float)n;
#pragma unroll
        for (int cc = 0; cc < 4; ++cc) {
            const int c = cc * 256 + tid;
            float s = Sl[il * PITCH + c];
            if (last) s -= x0r[c];
            const float xt = xw[il * PITCH + c];  // X_tilde[gi][c] (0 when gi==T)
            const size_t base = ((size_t)(b * (T_ + 1) + gi) * D_ + c) * (size_t)(H_ + 1);
#pragma unroll
            for (int h = 0; h < H_; ++h) {
                float v;
                if (h <= gi) {
                    const int gr = gi - h;
                    const float x = (gr >= t0) ? xw[(gr - t0) * PITCH + c]
                                               : X[(size_t)(b * T_ + gr) * D_ + c];
                    v = al * s + be * x;
                } else {
                    v = s * iu;
                }
                out[base + h] = v * inv[il * 8 + h];
            }
            out[base + 8] = xt;
        }
    }
}

// ---------------- Launch ----------------
extern "C" void kernel_launch(void* const* d_in, const int* in_sizes, int n_in,
                              void* d_out, int out_size, void* d_ws, size_t ws_size,
                              hipStream_t stream) {
    const float* X = (const float*)d_in[0];
    const float* W = (const float*)d_in[1];
    float* out = (float*)d_out;

    float* bs    = (float*)d_ws;                       // 4*128*1024 floats
    float* carry = bs + (size_t)B_ * NT_ * D_;         // 4*129*1024 floats

    k_blocksum<<<dim3(B_ * NT_), 256, 0, stream>>>(X, bs);
    k_carry<<<dim3(16), 256, 0, stream>>>(bs, carry);

    const size_t smem = (size_t)(2 * 16 * PITCH + D_ + 256 + 128) * sizeof(float);
    k_main<<<dim3(NTIL_, B_), 256, smem, stream>>>(X, W, carry, out);
}